// Spatial_CTX_16295105921034
// MI455X (gfx1250) — compile-verified
//
#include <hip/hip_runtime.h>
#include <hip/hip_bf16.h>
#include <stdint.h>

// ---------------------------------------------------------------------------
// Spatial context encoder: 12 grids (3x trilinear 3D + 9x bilinear 2D projections)
//   1) scatter-add weighted features + weights (global f32 atomics, L2-resident)
//   2) normalize wf / max(ws, 1e-9)
//   3) gather-interpolate M points: async global->LDS corner staging +
//      exact-f32 WMMA reduction (out16x32 = Wblockdiag(16x128) x V(128x32))
// ---------------------------------------------------------------------------

#define NGRIDS 12
#define FEAT 32
#define OUTCOLS (NGRIDS * FEAT)          // 384
#define TOTAL_CELLS 557056ULL            // sum of all grid cell counts

__constant__ int g_R[NGRIDS]  = {16, 32, 64,  64, 128, 256,  64, 128, 256,  64, 128, 256};
__constant__ int g_D[NGRIDS]  = {3, 3, 3,  2, 2, 2,  2, 2, 2,  2, 2, 2};
__constant__ int g_dA[NGRIDS] = {0, 0, 0,  0, 0, 0,  0, 0, 0,  1, 1, 1};
__constant__ int g_dB[NGRIDS] = {1, 1, 1,  1, 1, 1,  2, 2, 2,  2, 2, 2};
__constant__ long long g_cellOff[NGRIDS] = {
    0, 4096, 36864,
    299008, 303104, 319488,
    385024, 389120, 405504,
    471040, 475136, 491520};
__constant__ int g_colOff[NGRIDS] = {0, 32, 64, 96, 128, 160, 192, 224, 256, 288, 320, 352};

typedef float v2f __attribute__((ext_vector_type(2)));
typedef float v8f __attribute__((ext_vector_type(8)));

// Per-point corner weights + flat cell indices for grid g. Corners beyond 2^D
// get weight 0 / index 0 (index 0 is always a valid cell; weight 0 kills it).
__device__ __forceinline__ void corner_data(const float* __restrict__ xyz, int p, int g,
                                            float w8[8], int i8[8]) {
  const int D = g_D[g];
  const int R = g_R[g];
  float fr[3];
  int x0[3];
  for (int d = 0; d < 3; ++d) { fr[d] = 0.f; x0[d] = 0; }
  for (int d = 0; d < D; ++d) {
    int dim = (D == 3) ? d : ((d == 0) ? g_dA[g] : g_dB[g]);
    float x = xyz[(size_t)p * 3 + dim] * (float)(R - 1);
    float f0 = floorf(x);
    x0[d] = (int)f0;
    fr[d] = x - f0;
  }
  const int nc = 1 << D;
  for (int c = 0; c < 8; ++c) {
    if (c < nc) {
      float w = 1.f;
      int idx = 0;
      for (int d = 0; d < D; ++d) {
        int bit = (c >> d) & 1;
        int cc = x0[d] + bit;
        cc = cc < 0 ? 0 : (cc > R - 1 ? R - 1 : cc);
        w *= bit ? fr[d] : (1.f - fr[d]);
        idx = idx * R + cc;
      }
      w8[c] = w;
      i8[c] = idx;
    } else {
      w8[c] = 0.f;
      i8[c] = 0;
    }
  }
}

// ---------------------------------------------------------------------------
__global__ void zero_ws_kernel(float4* __restrict__ p, long long n4) {
  long long i = (long long)blockIdx.x * blockDim.x + threadIdx.x;
  if (i < n4) p[i] = make_float4(0.f, 0.f, 0.f, 0.f);
}

// ---------------------------------------------------------------------------
__global__ void scatter_kernel(const float* __restrict__ xyz,
                               const float* __restrict__ feat,
                               float* __restrict__ wf, float* __restrict__ ws, int N) {
  const int g = blockIdx.y;
  const int n = blockIdx.x * blockDim.x + threadIdx.x;
  if (n >= N) return;

  float w8[8];
  int i8[8];
  corner_data(xyz, n, g, w8, i8);

  float4 f[8];
  const float4* fp = (const float4*)(feat + (size_t)n * FEAT);
#pragma unroll
  for (int k = 0; k < 8; ++k) f[k] = fp[k];

  const long long cOff = g_cellOff[g];
  const int nc = 1 << g_D[g];
  for (int c = 0; c < nc; ++c) {
    const float w = w8[c];
    const size_t base = (size_t)(cOff + i8[c]);
    atomicAdd(ws + base, w);
    float* dst = wf + base * FEAT;
#pragma unroll
    for (int k = 0; k < 8; ++k) {
      atomicAdd(dst + 4 * k + 0, w * f[k].x);
      atomicAdd(dst + 4 * k + 1, w * f[k].y);
      atomicAdd(dst + 4 * k + 2, w * f[k].z);
      atomicAdd(dst + 4 * k + 3, w * f[k].w);
    }
  }
}

// ---------------------------------------------------------------------------
__global__ void normalize_kernel(float* __restrict__ wf, const float* __restrict__ ws) {
  size_t cid = (size_t)blockIdx.x * blockDim.x + threadIdx.x;
  if (cid >= TOTAL_CELLS) return;
  const float s = 1.0f / fmaxf(ws[cid], 1e-9f);
  float4* p = (float4*)(wf + cid * FEAT);
#pragma unroll
  for (int k = 0; k < 8; ++k) {
    float4 v = p[k];
    v.x *= s; v.y *= s; v.z *= s; v.w *= s;
    p[k] = v;
  }
}

// ---------------------------------------------------------------------------
// Encode: one wave32 handles one (grid g, 16-point tile).
//   phase1: lanes 0..15 compute 8 (w, idx) pairs for their point -> LDS
//   phase2: async global->LDS copy of 128 corner rows (128B each, coalesced)
//   phase3: out(16x32) = W(16x128) x V(128x32) via 64x V_WMMA_F32_16X16X4_F32
//           (W block-diagonal in f32 -> numerically exact weighted corner sum)
#define WPB 2  // waves per block
__global__ __launch_bounds__(32 * WPB) void encode_kernel(const float* __restrict__ xyz_i,
                                                          const float* __restrict__ wf,
                                                          float* __restrict__ out,
                                                          int M, int tilesPerGrid) {
  __shared__ float lds_v[WPB][128 * FEAT];  // 16KB per wave: 128 corner rows x 32 feats
  __shared__ float lds_w[WPB][16 * 8];
  __shared__ int lds_i[WPB][16 * 8];

  const int slot = threadIdx.x >> 5;
  const int lane = threadIdx.x & 31;
  const int tile = blockIdx.x * WPB + slot;
  const int g = tile / tilesPerGrid;     // wave-uniform
  const int t = tile - g * tilesPerGrid;
  const int m0 = t * 16;
  const bool active = (g < NGRIDS);

  // ---- phase 1: weights + indices ----
  if (active && lane < 16) {
    const int p = m0 + lane;
    float w8[8];
    int i8[8];
    if (p < M) {
      corner_data(xyz_i, p, g, w8, i8);
    } else {
      for (int c = 0; c < 8; ++c) { w8[c] = 0.f; i8[c] = 0; }
    }
    for (int c = 0; c < 8; ++c) {
      lds_w[slot][lane * 8 + c] = w8[c];
      lds_i[slot][lane * 8 + c] = i8[c];
    }
  }
  __syncthreads();

  // ---- phase 2: async gather of corner rows into LDS ----
  if (active) {
    const long long cOff = g_cellOff[g];
    const int group = lane >> 3;   // 4 rows per iteration
    const int f4 = lane & 7;       // 8 lanes x float4 = one 128B row
    for (int it = 0; it < 32; ++it) {
      const int r = it * 4 + group;           // row = point*8 + corner
      const int idx = lds_i[slot][r];
      const float* src = wf + (size_t)(cOff + idx) * FEAT + f4 * 4;
      unsigned ldsoff = (unsigned)(unsigned long long)&lds_v[slot][r * FEAT + f4 * 4];
      asm volatile("global_load_async_to_lds_b128 %0, %1, off"
                   :
                   : "v"(ldsoff), "v"(src)
                   : "memory");
    }
    asm volatile("s_wait_asynccnt 0" ::: "memory");
  }
  __syncthreads();

  // ---- phase 3: WMMA reduction ----
  if (active) {
    const int mrow = lane & 15;
    const int hi = lane >> 4;
    const int kb = hi * 2;  // A/B VGPR<->K map: lanes 0-15 hold K=0,1; 16-31 hold K=2,3
    const int colOff = g_colOff[g];

    // Lane L's A element is nonzero only for slices j with (j>>1)==mrow, and then
    // it is one of these 4 weights. Load them once; A becomes pure v_cndmask.
    const float2 wA0 = *(const float2*)&lds_w[slot][mrow * 8 + kb];      // corners 0..3
    const float2 wA1 = *(const float2*)&lds_w[slot][mrow * 8 + 4 + kb];  // corners 4..7

    v8f acc0 = {0.f, 0.f, 0.f, 0.f, 0.f, 0.f, 0.f, 0.f};
    v8f acc1 = {0.f, 0.f, 0.f, 0.f, 0.f, 0.f, 0.f, 0.f};

#pragma unroll
    for (int j = 0; j < 32; ++j) {          // K-slice j covers V rows 4j..4j+3
      const int p = j >> 1;                 // owning point of this slice
      const bool second = (j & 1) != 0;     // corner half (0..3 vs 4..7)
      const bool mine = (p == mrow);
      const float a0 = mine ? (second ? wA1.x : wA0.x) : 0.f;
      const float a1 = mine ? (second ? wA1.y : wA0.y) : 0.f;
      v2f A; A.x = a0; A.y = a1;

      const int rk = j * 4 + kb;
      // B (4x16) for both 16-column tiles; independent accumulators so the two
      // WMMAs pipeline back-to-back with no RAW hazard.
      v2f B0, B1;
      B0.x = lds_v[slot][rk * FEAT + mrow];
      B0.y = lds_v[slot][(rk + 1) * FEAT + mrow];
      B1.x = lds_v[slot][rk * FEAT + 16 + mrow];
      B1.y = lds_v[slot][(rk + 1) * FEAT + 16 + mrow];

      acc0 = __builtin_amdgcn_wmma_f32_16x16x4_f32(false, A, false, B0, (short)0, acc0,
                                                   false, false);
      acc1 = __builtin_amdgcn_wmma_f32_16x16x4_f32(false, A, false, B1, (short)0, acc1,
                                                   false, false);
    }

    // C/D layout: VGPR v -> row M = v + hi*8, col N = lane%16 (+ tile*16)
#pragma unroll
    for (int v = 0; v < 8; ++v) {
      const int mm = m0 + v + hi * 8;
      if (mm < M) {
        float* orow = out + (size_t)mm * OUTCOLS + colOff + mrow;
        orow[0] = acc0[v];
        orow[16] = acc1[v];
      }
    }
  }
}

// ---------------------------------------------------------------------------
extern "C" void kernel_launch(void* const* d_in, const int* in_sizes, int n_in,
                              void* d_out, int out_size, void* d_ws, size_t ws_size,
                              hipStream_t stream) {
  const float* xyz_c = (const float*)d_in[0];
  const float* xyz_i = (const float*)d_in[1];
  const float* feat = (const float*)d_in[2];
  const int N = in_sizes[0] / 3;
  const int M = in_sizes[1] / 3;

  float* wf = (float*)d_ws;                            // TOTAL_CELLS * 32 floats
  float* ws = wf + TOTAL_CELLS * (size_t)FEAT;         // TOTAL_CELLS floats
  float* out = (float*)d_out;

  // zero wf + ws (18,382,848 floats, divisible by 4)
  const long long n4 = (long long)(TOTAL_CELLS * 33ULL / 4ULL);
  zero_ws_kernel<<<(unsigned)((n4 + 255) / 256), 256, 0, stream>>>((float4*)d_ws, n4);

  // scatter: one thread per creator point, one grid per blockIdx.y
  dim3 sgrid((N + 255) / 256, NGRIDS);
  scatter_kernel<<<sgrid, 256, 0, stream>>>(xyz_c, feat, wf, ws, N);

  // normalize
  normalize_kernel<<<(unsigned)((TOTAL_CELLS + 255) / 256), 256, 0, stream>>>(wf, ws);

  // encode: one wave per (grid, 16-point tile)
  const int tpg = (M + 15) / 16;
  const long long totalTiles = (long long)NGRIDS * tpg;
  const unsigned eblocks = (unsigned)((totalTiles + WPB - 1) / WPB);
  encode_kernel<<<eblocks, 32 * WPB, 0, stream>>>(xyz_i, wf, out, M, tpg);
}